// Conv2D_52991306498149
// MI455X (gfx1250) — compile-verified
//
#include <hip/hip_runtime.h>
#include <hip/hip_bf16.h>

#define H 8192
#define W 8192
#define OH 8190
#define OW 8190

#define TILE_X 256          // output cols per block
#define TILE_Y 32           // output rows per block
#define IN_ROWS (TILE_Y + 2)   // 34 input rows staged
#define CHUNKS_X 65            // 65 * 4 floats = 260 >= 258 needed cols
#define LDSW 272               // LDS row stride in floats (16B aligned, padded)

#define GLOBAL_AS __attribute__((address_space(1)))
#define LDS_AS    __attribute__((address_space(3)))

typedef int v4i __attribute__((ext_vector_type(4)));
typedef GLOBAL_AS v4i* gv4i_p;   // global-addrspace pointer to 16B vector
typedef LDS_AS    v4i* lv4i_p;   // LDS-addrspace pointer to 16B vector

__device__ __forceinline__ void async_load_b128(const float* g, float* l) {
#if defined(__has_builtin) && __has_builtin(__builtin_amdgcn_global_load_async_to_lds_b128)
    // CDNA5 async DMA: global -> LDS, 16 bytes per lane, tracked by ASYNCcnt
    __builtin_amdgcn_global_load_async_to_lds_b128(
        (gv4i_p)g, (lv4i_p)l, /*offset=*/0, /*cpol=*/0);
#else
    *(float4*)l = *(const float4*)g;   // synchronous fallback
#endif
}

__device__ __forceinline__ void wait_async_lds() {
#if defined(__has_builtin) && __has_builtin(__builtin_amdgcn_s_wait_asynccnt)
    __builtin_amdgcn_s_wait_asynccnt(0);
#else
    asm volatile("s_wait_asynccnt 0" ::: "memory");
#endif
}

__global__ __launch_bounds__(256)
void Conv2D_3x3_kernel(const float* __restrict__ in,
                       const float* __restrict__ wgt,
                       const float* __restrict__ bias,
                       float* __restrict__ out)
{
    __shared__ __align__(16) float smem[IN_ROWS * LDSW];

    const int t  = threadIdx.x;
    const int x0 = blockIdx.x * TILE_X;   // first output col of tile
    const int y0 = blockIdx.y * TILE_Y;   // first output row of tile

    // ---- weights / bias into registers (uniform, tiny, L2-cached) ----
    const float w00 = wgt[0], w01 = wgt[1], w02 = wgt[2];
    const float w10 = wgt[3], w11 = wgt[4], w12 = wgt[5];
    const float w20 = wgt[6], w21 = wgt[7], w22 = wgt[8];
    const float bv  = bias[0];

    // ---- async stage input tile: 34 rows x 260 cols (65 x b128 per row) ----
    for (int c = t; c < IN_ROWS * CHUNKS_X; c += 256) {
        const int row  = c / CHUNKS_X;
        const int col4 = (c - row * CHUNKS_X) * 4;
        const int gy = y0 + row;
        const int gx = x0 + col4;
        if (gy < H && gx < W) {   // gx multiple of 4 and < W  => gx+3 <= W-1
            async_load_b128(&in[(size_t)gy * W + gx],
                            &smem[row * LDSW + col4]);
        }
    }
    wait_async_lds();
    __syncthreads();

    // ---- compute: each thread -> 8 rows x 4 cols, rolling 3-row window ----
    const int tx = t & 63;        // 0..63  -> col group
    const int ty = t >> 6;        // 0..3   -> row group
    const int c0 = tx * 4;        // col within tile
    const int r0 = ty * 8;        // first output row within tile

    float buf[3][6];

    auto load6 = [&](int rrow, float* dst) {
        const float* p = &smem[rrow * LDSW + c0];   // 16B aligned (c0 % 4 == 0)
        const float4 a = *(const float4*)p;
        const float2 b = *(const float2*)(p + 4);
        dst[0] = a.x; dst[1] = a.y; dst[2] = a.z; dst[3] = a.w;
        dst[4] = b.x; dst[5] = b.y;
    };

    load6(r0 + 0, buf[0]);
    load6(r0 + 1, buf[1]);

#pragma unroll
    for (int j = 0; j < 8; ++j) {
        load6(r0 + j + 2, buf[(j + 2) % 3]);
        const float* t0 = buf[(j + 0) % 3];
        const float* t1 = buf[(j + 1) % 3];
        const float* t2 = buf[(j + 2) % 3];

        float o[4];
#pragma unroll
        for (int i = 0; i < 4; ++i) {
            float acc = bv;
            acc = fmaf(w00, t0[i],     acc);
            acc = fmaf(w01, t0[i + 1], acc);
            acc = fmaf(w02, t0[i + 2], acc);
            acc = fmaf(w10, t1[i],     acc);
            acc = fmaf(w11, t1[i + 1], acc);
            acc = fmaf(w12, t1[i + 2], acc);
            acc = fmaf(w20, t2[i],     acc);
            acc = fmaf(w21, t2[i + 1], acc);
            acc = fmaf(w22, t2[i + 2], acc);
            o[i] = acc;
        }

        const int oy = y0 + r0 + j;
        const int ox = x0 + c0;
        if (oy < OH) {
            // byte address always 8-aligned: (oy*8190 + ox) is even
            const size_t base = (size_t)oy * OW + ox;
            if (ox + 1 < OW) *(float2*)&out[base]     = make_float2(o[0], o[1]);
            if (ox + 3 < OW) *(float2*)&out[base + 2] = make_float2(o[2], o[3]);
        }
    }
}

extern "C" void kernel_launch(void* const* d_in, const int* in_sizes, int n_in,
                              void* d_out, int out_size, void* d_ws, size_t ws_size,
                              hipStream_t stream) {
    const float* x = (const float*)d_in[0];   // 8192*8192 fp32
    const float* w = (const float*)d_in[1];   // 3*3 fp32
    const float* b = (const float*)d_in[2];   // 1 fp32
    float* out = (float*)d_out;               // 8190*8190 fp32

    dim3 block(256, 1, 1);
    dim3 grid((OW + TILE_X - 1) / TILE_X,    // 32
              (OH + TILE_Y - 1) / TILE_Y,    // 256
              1);
    Conv2D_3x3_kernel<<<grid, block, 0, stream>>>(x, w, b, out);
}